// NumbaNeighborList_1614907703798
// MI455X (gfx1250) — compile-verified
//
#include <hip/hip_runtime.h>
#include <hip/hip_bf16.h>

#define TILE 256
#define CUTOFF2 36.0f  // 6.0^2

// ---------------------------------------------------------------------------
// Dense padded neighbor list with PBC (minimum image), box > 2*cutoff.
// Outputs (concatenated flat in d_out, all as float32):
//   [0)            : Rij_m   [N][N][3]
//   [3*N*N)        : offsets [N][N][3]  (integer-valued shifts, masked)
//   [6*N*N)        : mask    [N][N]     (1.0 / 0.0)
// Grid: (N/TILE j-tiles, N rows). Block: 256 threads = 8 wave32s.
// Each block stages its 256-atom j-tile into LDS via the CDNA5 async
// global->LDS path; outputs are streamed with non-temporal store hints
// (470 MB write stream > 192 MB L2, never re-read by the kernel).
// ---------------------------------------------------------------------------

typedef __attribute__((address_space(1))) int gint_t;  // global
typedef __attribute__((address_space(3))) int lint_t;  // LDS

__global__ __launch_bounds__(TILE) void nbrlist_kernel(
    const float* __restrict__ pos,          // [N*3]
    const float* __restrict__ cell,         // [9], diagonal box
    const unsigned char* __restrict__ pbc,  // [3] bools
    float* __restrict__ outR,               // [N*N*3]
    float* __restrict__ outOff,             // [N*N*3]
    float* __restrict__ outMask,            // [N*N]
    int n)
{
    __shared__ float sj[TILE * 3];

    const int tid      = threadIdx.x;
    const int i        = blockIdx.y;
    const int tileBase = blockIdx.x * TILE;
    const int tileCnt  = min(TILE, n - tileBase);
    const int ndw      = tileCnt * 3;  // dwords to stage into LDS

    // ---- stage j-tile positions into LDS (CDNA5 async path) ----
    const float* gsrc = pos + (size_t)tileBase * 3;
#if defined(__gfx1250__) && __has_builtin(__builtin_amdgcn_global_load_async_to_lds_b32)
    for (int k = tid; k < ndw; k += TILE) {
        __builtin_amdgcn_global_load_async_to_lds_b32(
            (gint_t*)(gsrc + k),
            (lint_t*)(sj + k),
            /*offset=*/0, /*cpol=*/0);
    }
  #if __has_builtin(__builtin_amdgcn_s_wait_asynccnt)
    __builtin_amdgcn_s_wait_asynccnt(0);
  #else
    asm volatile("s_wait_asynccnt 0" ::: "memory");
  #endif
#else
    for (int k = tid; k < ndw; k += TILE) sj[k] = gsrc[k];
#endif
    __syncthreads();

    // ---- per-pair compute ----
    const float bx = cell[0];
    const float by = cell[4];
    const float bz = cell[8];
    const bool  p0 = pbc[0] != 0;
    const bool  p1 = pbc[1] != 0;
    const bool  p2 = pbc[2] != 0;

    const float pix = pos[(size_t)i * 3 + 0];
    const float piy = pos[(size_t)i * 3 + 1];
    const float piz = pos[(size_t)i * 3 + 2];

    const int j = tileBase + tid;
    if (j >= n) return;

    const float pjx = sj[tid * 3 + 0];
    const float pjy = sj[tid * 3 + 1];
    const float pjz = sj[tid * 3 + 2];

    // diff = r_j - r_i
    float dx = pjx - pix;
    float dy = pjy - piy;
    float dz = pjz - piz;

    // minimum-image shift: -round(diff/box) (RTNE == jnp.round), 0 if not periodic
    float sx = p0 ? -rintf(dx / bx) : 0.0f;
    float sy = p1 ? -rintf(dy / by) : 0.0f;
    float sz = p2 ? -rintf(dz / bz) : 0.0f;

    float rx = fmaf(sx, bx, dx);
    float ry = fmaf(sy, by, dy);
    float rz = fmaf(sz, bz, dz);

    const float d2 = rx * rx + ry * ry + rz * rz;
    const bool  m  = (d2 < CUTOFF2) && (j != i);
    const float mm = m ? 1.0f : 0.0f;

    // masked outputs
    rx *= mm;  ry *= mm;  rz *= mm;
    const float ox = m ? sx : 0.0f;
    const float oy = m ? sy : 0.0f;
    const float oz = m ? sz : 0.0f;

    const size_t pair = (size_t)i * (size_t)n + (size_t)j;
    const size_t b3   = pair * 3;

    // Streaming outputs: non-temporal hint keeps the 470 MB write stream
    // from thrashing L2. 12 consecutive bytes per thread, contiguous across
    // lanes -> coalesced.
    __builtin_nontemporal_store(rx, &outR[b3 + 0]);
    __builtin_nontemporal_store(ry, &outR[b3 + 1]);
    __builtin_nontemporal_store(rz, &outR[b3 + 2]);

    __builtin_nontemporal_store(ox, &outOff[b3 + 0]);
    __builtin_nontemporal_store(oy, &outOff[b3 + 1]);
    __builtin_nontemporal_store(oz, &outOff[b3 + 2]);

    __builtin_nontemporal_store(mm, &outMask[pair]);
}

extern "C" void kernel_launch(void* const* d_in, const int* in_sizes, int n_in,
                              void* d_out, int out_size, void* d_ws, size_t ws_size,
                              hipStream_t stream) {
    const float*         pos  = (const float*)d_in[0];          // [N,3]
    const float*         cell = (const float*)d_in[1];          // [3,3]
    const unsigned char* pbc  = (const unsigned char*)d_in[2];  // [3] bool

    const int n = in_sizes[0] / 3;

    float* outR    = (float*)d_out;
    float* outOff  = outR   + (size_t)n * (size_t)n * 3;
    float* outMask = outOff + (size_t)n * (size_t)n * 3;

    dim3 block(TILE);
    dim3 grid((n + TILE - 1) / TILE, n);
    hipLaunchKernelGGL(nbrlist_kernel, grid, block, 0, stream,
                       pos, cell, pbc, outR, outOff, outMask, n);
}